// SRR_CG_85658827752045
// MI455X (gfx1250) — compile-verified
//
#include <hip/hip_runtime.h>
#include <hip/hip_bf16.h>

#define D_   128
#define H_   128
#define W_   128
#define NSL  16
#define HS_  128
#define WS_  128
#define K_   27
#define NV   (D_ * H_ * W_)        // 2,097,152
#define NPIX (NSL * HS_ * WS_)     // 262,144
#define RESR 1.5f
#define NITER 10
#define NVBLK (NV / 256)           // 8192

typedef __attribute__((ext_vector_type(2))) float v2f;
typedef __attribute__((ext_vector_type(8))) float v8f;

// ---------------------------------------------------------------------------
// Wave-wide fp32 reduction through the CDNA5 matrix pipe.
// Each of the 32 lanes contributes `partial` once into an A(16x4) fragment
// (other slot zero).  B(4x16) = all ones, so D[m][n] = rowsum(A) for every n.
// Summing the 8 C VGPRs per lane yields sum over M=0..7 (lanes 0-15) and
// M=8..15 (lanes 16-31); one shfl_xor(16) completes the full wave sum.
// Correct regardless of which (m,k) slot a given lane's value maps to.
// Requires EXEC == all ones: callers launch exact grids, no divergence here.
// ---------------------------------------------------------------------------
__device__ __forceinline__ float wave_reduce_wmma(float partial) {
  v2f a; a[0] = partial; a[1] = 0.0f;
  v2f b; b[0] = 1.0f;    b[1] = 1.0f;
  v8f c = {};
  c = __builtin_amdgcn_wmma_f32_16x16x4_f32(
      /*neg_a=*/false, a, /*neg_b=*/false, b,
      /*c_mod=*/(short)0, c, /*reuse_a=*/false, /*reuse_b=*/false);
  float s = ((c[0] + c[1]) + (c[2] + c[3])) + ((c[4] + c[5]) + (c[6] + c[7]));
  s += __shfl_xor(s, 16, 32);
  return s;
}

// ---------------------------------------------------------------------------
// Precompute rotated PSF offsets:  roffs[n][k][r] = sum_c offs[k][c] * R[n][r][c]
// offs[k] = (kx-1, ky-1, kz-1), k = kz*9 + ky*3 + kx  (matches jnp.meshgrid 'ij')
// ---------------------------------------------------------------------------
__global__ void k_roffs(const float* __restrict__ theta, float* __restrict__ roffs) {
  int t = blockIdx.x * blockDim.x + threadIdx.x;
  if (t >= NSL * K_) return;
  int n = t / K_, k = t % K_;
  float ox = (float)(k % 3) - 1.0f;
  float oy = (float)((k / 3) % 3) - 1.0f;
  float oz = (float)(k / 9) - 1.0f;
  const float* th = theta + n * 12;
  roffs[t * 3 + 0] = ox * th[0] + oy * th[1] + oz * th[2];
  roffs[t * 3 + 1] = ox * th[4] + oy * th[5] + oz * th[6];
  roffs[t * 3 + 2] = ox * th[8] + oy * th[9] + oz * th[10];
}

__global__ void k_fill_zero(float* __restrict__ dst) {
  dst[blockIdx.x * 256 + threadIdx.x] = 0.0f;
}

__global__ void k_zero_sc(float* __restrict__ sc) {
  if (threadIdx.x < 64) sc[threadIdx.x] = 0.0f;
}

// ---------------------------------------------------------------------------
// A_op: PSF-weighted trilinear gather, volume -> slices.
// One thread per slice pixel.  64 blocks per slice => uniform (R,t) per block.
// ---------------------------------------------------------------------------
__global__ void k_gather(const float* __restrict__ theta,
                         const float* __restrict__ roffs,
                         const float* __restrict__ psf,
                         const float* __restrict__ vol,
                         float* __restrict__ slc) {
  int bx  = blockIdx.x;
  int n   = bx >> 6;                       // 64 blocks of 256 per slice
  int pix = ((bx & 63) << 8) | threadIdx.x;
  int h = pix >> 7, w = pix & 127;

  const float* th = theta + n * 12;
  float R00 = th[0], R01 = th[1], tx = th[3];
  float R10 = th[4], R11 = th[5], ty = th[7];
  float R20 = th[8], R21 = th[9], tz = th[11];

  float u = ((float)w - 63.5f) * RESR;
  float v = ((float)h - 63.5f) * RESR;
  float p0x = u * R00 + v * R01 + tx + 63.5f;
  float p0y = u * R10 + v * R11 + ty + 63.5f;
  float p0z = u * R20 + v * R21 + tz + 63.5f;

  // warm L2/L0 for this pixel's neighborhood (global_prefetch_b8)
  {
    int cx = min(max((int)p0x, 0), W_ - 1);
    int cy = min(max((int)p0y, 0), H_ - 1);
    int cz = min(max((int)p0z, 0), D_ - 1);
    __builtin_prefetch(vol + ((((cz << 7) + cy) << 7) + cx), 0, 0);
  }

  const float* ro = roffs + n * (K_ * 3);
  float acc = 0.0f;
  for (int k = 0; k < K_; ++k) {
    float px = p0x + ro[k * 3 + 0];
    float py = p0y + ro[k * 3 + 1];
    float pz = p0z + ro[k * 3 + 2];
    float fx0 = floorf(px), fy0 = floorf(py), fz0 = floorf(pz);
    int ix = (int)fx0, iy = (int)fy0, iz = (int)fz0;
    float fx = px - fx0, fy = py - fy0, fz = pz - fz0;
    float wx[2] = {1.0f - fx, fx};
    float wy[2] = {1.0f - fy, fy};
    float wz[2] = {1.0f - fz, fz};
    float acck = 0.0f;
#pragma unroll
    for (int dz = 0; dz < 2; ++dz)
#pragma unroll
      for (int dy = 0; dy < 2; ++dy)
#pragma unroll
        for (int dx = 0; dx < 2; ++dx) {
          int xi = ix + dx, yi = iy + dy, zi = iz + dz;
          bool inb = ((unsigned)xi < (unsigned)W_) &
                     ((unsigned)yi < (unsigned)H_) &
                     ((unsigned)zi < (unsigned)D_);
          int lin   = inb ? ((((zi << 7) + yi) << 7) + xi) : 0;
          float wgt = inb ? (wx[dx] * wy[dy] * wz[dz]) : 0.0f;
          acck = fmaf(vol[lin], wgt, acck);
        }
    acc = fmaf(acck, psf[k], acc);
  }
  slc[n * (HS_ * WS_) + pix] = acc;
}

// ---------------------------------------------------------------------------
// At_op: PSF-weighted trilinear scatter-add, slices -> volume (atomics in L2).
// ---------------------------------------------------------------------------
__global__ void k_scatter(const float* __restrict__ theta,
                          const float* __restrict__ roffs,
                          const float* __restrict__ psf,
                          const float* __restrict__ slc,
                          float* __restrict__ vol) {
  int bx  = blockIdx.x;
  int n   = bx >> 6;
  int pix = ((bx & 63) << 8) | threadIdx.x;
  int h = pix >> 7, w = pix & 127;

  const float* th = theta + n * 12;
  float R00 = th[0], R01 = th[1], tx = th[3];
  float R10 = th[4], R11 = th[5], ty = th[7];
  float R20 = th[8], R21 = th[9], tz = th[11];

  float u = ((float)w - 63.5f) * RESR;
  float v = ((float)h - 63.5f) * RESR;
  float p0x = u * R00 + v * R01 + tx + 63.5f;
  float p0y = u * R10 + v * R11 + ty + 63.5f;
  float p0z = u * R20 + v * R21 + tz + 63.5f;

  float sval = slc[n * (HS_ * WS_) + pix];
  const float* ro = roffs + n * (K_ * 3);

  for (int k = 0; k < K_; ++k) {
    float wv = sval * psf[k];
    float px = p0x + ro[k * 3 + 0];
    float py = p0y + ro[k * 3 + 1];
    float pz = p0z + ro[k * 3 + 2];
    float fx0 = floorf(px), fy0 = floorf(py), fz0 = floorf(pz);
    int ix = (int)fx0, iy = (int)fy0, iz = (int)fz0;
    float fx = px - fx0, fy = py - fy0, fz = pz - fz0;
    float wx[2] = {1.0f - fx, fx};
    float wy[2] = {1.0f - fy, fy};
    float wz[2] = {1.0f - fz, fz};
#pragma unroll
    for (int dz = 0; dz < 2; ++dz)
#pragma unroll
      for (int dy = 0; dy < 2; ++dy)
#pragma unroll
        for (int dx = 0; dx < 2; ++dx) {
          int xi = ix + dx, yi = iy + dy, zi = iz + dz;
          bool inb = ((unsigned)xi < (unsigned)W_) &
                     ((unsigned)yi < (unsigned)H_) &
                     ((unsigned)zi < (unsigned)D_);
          if (inb) {
            int lin = (((zi << 7) + yi) << 7) + xi;
            atomicAdd(&vol[lin], wx[dx] * wy[dy] * wz[dz] * wv);
          }
        }
  }
}

// dot(a,b) over NV elements -> sc[slot]; exact grid, full EXEC for WMMA.
__global__ void k_dot(const float* __restrict__ a, const float* __restrict__ b,
                      float* __restrict__ sc, int slot) {
  int i = blockIdx.x * 256 + threadIdx.x;
  float s = wave_reduce_wmma(a[i] * b[i]);
  if ((threadIdx.x & 31) == 0) atomicAdd(&sc[slot], s);
}

// x = x0; r = b - Ap; p = r; sc[0] += dot(r,r).
// NOTE: b and x intentionally alias (workspace reuse) => NO __restrict__ on
// them, and the b load is forced to complete before the x store.
__global__ void k_init(const float* __restrict__ vol0,
                       const float* b,
                       const float* __restrict__ Ap,
                       float* x, float* __restrict__ r,
                       float* __restrict__ p, float* __restrict__ sc) {
  int i = blockIdx.x * 256 + threadIdx.x;
  float bi = b[i];              // read b before writing x (same buffer)
  float ri = bi - Ap[i];
  x[i] = vol0[i];
  r[i] = ri;
  p[i] = ri;
  float s = wave_reduce_wmma(ri * ri);
  if ((threadIdx.x & 31) == 0) atomicAdd(&sc[0], s);
}

// alpha = sc[rr]/sc[pap]; x += alpha p; r -= alpha Ap; sc[rrn] += dot(r,r)
__global__ void k_update_xr(float* __restrict__ x, float* __restrict__ r,
                            const float* __restrict__ p, const float* __restrict__ Ap,
                            float* __restrict__ sc, int rr_slot, int pap_slot,
                            int rrn_slot) {
  int i = blockIdx.x * 256 + threadIdx.x;
  float alpha = sc[rr_slot] / sc[pap_slot];
  x[i] = fmaf(alpha, p[i], x[i]);
  float rn = fmaf(-alpha, Ap[i], r[i]);
  r[i] = rn;
  float s = wave_reduce_wmma(rn * rn);
  if ((threadIdx.x & 31) == 0) atomicAdd(&sc[rrn_slot], s);
}

// beta = sc[rrn]/sc[rr]; p = r + beta*p
__global__ void k_update_p(float* __restrict__ p, const float* __restrict__ r,
                           const float* __restrict__ sc, int rrn_slot, int rr_slot) {
  int i = blockIdx.x * 256 + threadIdx.x;
  float beta = sc[rrn_slot] / sc[rr_slot];
  p[i] = fmaf(beta, p[i], r[i]);
}

__global__ void k_relu(const float* __restrict__ x, float* __restrict__ out) {
  int i = blockIdx.x * 256 + threadIdx.x;
  out[i] = fmaxf(x[i], 0.0f);
}

// ---------------------------------------------------------------------------
extern "C" void kernel_launch(void* const* d_in, const int* in_sizes, int n_in,
                              void* d_out, int out_size, void* d_ws, size_t ws_size,
                              hipStream_t stream) {
  (void)in_sizes; (void)n_in; (void)out_size; (void)ws_size;
  const float* theta  = (const float*)d_in[0];   // [16,3,4]
  const float* slices = (const float*)d_in[1];   // [16,1,128,128]
  const float* vol0   = (const float*)d_in[2];   // [1,1,128,128,128]
  const float* psf    = (const float*)d_in[3];   // [3,3,3]
  float* out = (float*)d_out;

  // workspace layout (floats); b aliases X (consumed exactly where x is first written)
  float* ws = (float*)d_ws;
  float* X  = ws;                        // NV   (also serves as b during setup)
  float* R  = ws + (size_t)NV;           // NV
  float* P  = ws + 2 * (size_t)NV;       // NV
  float* AP = ws + 3 * (size_t)NV;       // NV
  float* S  = ws + 4 * (size_t)NV;       // NPIX  slice scratch
  float* RO = S + NPIX;                  // 16*27*3 = 1296
  float* SC = RO + 2048;                 // 64 scalar slots

  const dim3 blk(256);
  const dim3 gPix(NPIX / 256);           // 1024 (64 blocks per slice)
  const dim3 gVol(NVBLK);                // 8192

  k_roffs<<<dim3(2), blk, 0, stream>>>(theta, RO);
  k_zero_sc<<<dim3(1), dim3(64), 0, stream>>>(SC);

  // b = At(slices)    (stored in X region)
  k_fill_zero<<<gVol, blk, 0, stream>>>(X);
  k_scatter<<<gPix, blk, 0, stream>>>(theta, RO, psf, slices, X);

  // Ap = AtA(x0)
  k_gather<<<gPix, blk, 0, stream>>>(theta, RO, psf, vol0, S);
  k_fill_zero<<<gVol, blk, 0, stream>>>(AP);
  k_scatter<<<gPix, blk, 0, stream>>>(theta, RO, psf, S, AP);

  // x = x0; r = p = b - Ap; sc[0] = rr0
  k_init<<<gVol, blk, 0, stream>>>(vol0, X /*b*/, AP, X, R, P, SC);

  for (int i = 0; i < NITER; ++i) {
    int rr_prev = (i == 0) ? 0 : 2 * i;
    int pap     = 1 + 2 * i;
    int rrn     = 2 + 2 * i;
    k_gather<<<gPix, blk, 0, stream>>>(theta, RO, psf, P, S);
    k_fill_zero<<<gVol, blk, 0, stream>>>(AP);
    k_scatter<<<gPix, blk, 0, stream>>>(theta, RO, psf, S, AP);
    k_dot<<<gVol, blk, 0, stream>>>(P, AP, SC, pap);
    k_update_xr<<<gVol, blk, 0, stream>>>(X, R, P, AP, SC, rr_prev, pap, rrn);
    k_update_p<<<gVol, blk, 0, stream>>>(P, R, SC, rrn, rr_prev);
  }

  k_relu<<<gVol, blk, 0, stream>>>(X, out);
}